// BWGNN_63101659513087
// MI455X (gfx1250) — compile-verified
//
#include <hip/hip_runtime.h>

typedef __attribute__((ext_vector_type(2))) float v2f;
typedef __attribute__((ext_vector_type(8))) float v8f;

// ---------------- utility kernels ----------------

__global__ void zero_f32(float* __restrict__ p, int n) {
    int i = blockIdx.x * blockDim.x + threadIdx.x;
    if (i < n) p[i] = 0.0f;
}

__global__ void deg_kernel(const int* __restrict__ dst, float* __restrict__ deg, int n_edges) {
    int e = blockIdx.x * blockDim.x + threadIdx.x;
    if (e < n_edges) atomicAdd(&deg[dst[e]], 1.0f);
}

__global__ void dinv_kernel(float* __restrict__ deg, int n_nodes) {
    int v = blockIdx.x * blockDim.x + threadIdx.x;
    if (v < n_nodes) {
        float d = deg[v];
        d = d < 1.0f ? 1.0f : d;
        deg[v] = rsqrtf(d);
    }
}

// agg[dst] += (feat[src] * dinv[src]); 16 threads per edge, float4 each
__global__ void scatter_kernel(const int* __restrict__ src, const int* __restrict__ dst,
                               const float* __restrict__ feat, const float* __restrict__ dinv,
                               float* __restrict__ agg, int n_edges) {
    int t = blockIdx.x * blockDim.x + threadIdx.x;
    int e = t >> 4;
    if (e >= n_edges) return;
    int c = (t & 15) * 4;
    int s = src[e];
    int d = dst[e];
    float ds = dinv[s];
    const float4 v = *(const float4*)(feat + (size_t)s * 64 + c);
    float* o = agg + (size_t)d * 64 + c;
    atomicAdd(o + 0, v.x * ds);
    atomicAdd(o + 1, v.y * ds);
    atomicAdd(o + 2, v.z * ds);
    atomicAdd(o + 3, v.w * ds);
}

// feat_next = feat - agg * dinv[node]
__global__ void update_kernel(const float* __restrict__ feat, const float* __restrict__ agg,
                              const float* __restrict__ dinv, float* __restrict__ out, int n_nodes) {
    int t = blockIdx.x * blockDim.x + threadIdx.x;  // one float4 per thread
    int v = t >> 4;
    if (v >= n_nodes) return;
    int c = (t & 15) * 4;
    float dv = dinv[v];
    size_t base = (size_t)v * 64 + c;
    float4 f = *(const float4*)(feat + base);
    float4 a = *(const float4*)(agg + base);
    float4 r;
    r.x = f.x - a.x * dv;
    r.y = f.y - a.y * dv;
    r.z = f.z - a.z * dv;
    r.w = f.w - a.w * dv;
    *(float4*)(out + base) = r;
}

// In-place theta combine:
//   H  <- 3*H - 3*F1 + 0.75*F2   (out0, fed to W3)
//   F1 <- 3*F1 - 1.5*F2          (out1, low half of concat fed to W4)
//   F2 <- 0.75*F2                (out2, high half of concat fed to W4)
__global__ void combine_kernel(float* __restrict__ H, float* __restrict__ F1,
                               float* __restrict__ F2, int n_elems) {
    int i = blockIdx.x * blockDim.x + threadIdx.x;
    if (i >= n_elems) return;
    float h = H[i], f1 = F1[i], f2 = F2[i];
    H[i]  = 3.0f * h - 3.0f * f1 + 0.75f * f2;
    F1[i] = 3.0f * f1 - 1.5f * f2;
    F2[i] = 0.75f * f2;
}

// ---------------- WMMA fp32 GEMM:  out = relu(A @ W + b) ----------------
// A: n_rows x K (row major).  If SPLIT: A[m][k] = (k<64 ? A0[m*64+k] : A1[m*64+k-64]).
// W: K x 64 row-major, staged into LDS in K-pair-interleaved layout so each B
// fragment {W[k0][n], W[k0+1][n]} is one aligned ds_load_b64:
//   Wl[(k>>1)*128 + n*2 + (k&1)] = W[k*64 + n]
// One wave computes a 16x64 tile via 4 accumulators of V_WMMA_F32_16X16X4_F32.
template <int K, bool SPLIT>
__global__ void gemm_relu_wmma(const float* __restrict__ A0, const float* __restrict__ A1,
                               const float* __restrict__ W, const float* __restrict__ bias,
                               float* __restrict__ out, int n_rows) {
    __shared__ __align__(16) float Wlds[K * 64];
    for (int i = threadIdx.x; i < K * 64; i += blockDim.x) {
        int k = i >> 6;
        int n = i & 63;
        Wlds[(k >> 1) * 128 + n * 2 + (k & 1)] = W[i];
    }
    __syncthreads();

    const int lane = threadIdx.x & 31;
    const int wave = threadIdx.x >> 5;
    const int tile = blockIdx.x * (blockDim.x >> 5) + wave;  // 16-row tile index
    const int m0 = tile * 16;
    if (m0 >= n_rows) return;

    const int half = lane >> 4;  // selects K-pair (A/B) or row-group (C/D)
    const int mn = lane & 15;    // A row within tile / B,C,D column

    int arow = m0 + mn;
    if (arow >= n_rows) arow = n_rows - 1;  // safety clamp (N%16==0 -> unused)

    v8f acc[4] = {v8f{}, v8f{}, v8f{}, v8f{}};

    for (int kb = 0; kb < K; kb += 4) {
        const int k0 = kb + 2 * half;  // k0 even; k0 and k0+1 live in same sub-buffer
        v2f a;
        if (!SPLIT) {
            a = *(const v2f*)(A0 + (size_t)arow * K + k0);
        } else {
            const float* Ar = (k0 < 64) ? (A0 + (size_t)arow * 64 + k0)
                                        : (A1 + (size_t)arow * 64 + (k0 - 64));
            a = *(const v2f*)Ar;
        }
        const float* Wrow = Wlds + (size_t)(k0 >> 1) * 128 + mn * 2;
#pragma unroll
        for (int nt = 0; nt < 4; ++nt) {
            v2f b = *(const v2f*)(Wrow + nt * 32);  // single ds_load_b64
            acc[nt] = __builtin_amdgcn_wmma_f32_16x16x4_f32(
                false, a, false, b, (short)0, acc[nt], false, false);
        }
    }

    // Epilogue: bias + relu + store.  D layout: VGPR r -> row r + 8*half, col = mn.
    // Hot path: tile fully in range (always true when n_rows % 16 == 0) ->
    // branch-free plain stores, no per-element exec-mask churn.
    const bool full = (m0 + 16 <= n_rows);
    if (full) {
#pragma unroll
        for (int nt = 0; nt < 4; ++nt) {
            const int n = nt * 16 + mn;
            const float bn = bias[n];
            float* o = out + (size_t)(m0 + 8 * half) * 64 + n;
#pragma unroll
            for (int r = 0; r < 8; ++r) {
                float v = acc[nt][r] + bn;
                o[(size_t)r * 64] = v > 0.0f ? v : 0.0f;
            }
        }
    } else {
#pragma unroll
        for (int nt = 0; nt < 4; ++nt) {
            const int n = nt * 16 + mn;
            const float bn = bias[n];
#pragma unroll
            for (int r = 0; r < 8; ++r) {
                const int m = m0 + r + 8 * half;
                if (m < n_rows) {
                    float v = acc[nt][r] + bn;
                    out[(size_t)m * 64 + n] = v > 0.0f ? v : 0.0f;
                }
            }
        }
    }
}

// ---------------- host launch ----------------

extern "C" void kernel_launch(void* const* d_in, const int* in_sizes, int n_in,
                              void* d_out, int out_size, void* d_ws, size_t ws_size,
                              hipStream_t stream) {
    const float* in_feat = (const float*)d_in[0];
    const int* src = (const int*)d_in[1];
    const int* dst = (const int*)d_in[2];
    const float* W1 = (const float*)d_in[3];
    const float* b1 = (const float*)d_in[4];
    const float* W2 = (const float*)d_in[5];
    const float* b2 = (const float*)d_in[6];
    const float* W3 = (const float*)d_in[7];
    const float* b3 = (const float*)d_in[8];
    const float* W4 = (const float*)d_in[9];
    const float* b4 = (const float*)d_in[10];

    const int n_nodes = in_sizes[0] / 128;  // 100000
    const int n_edges = in_sizes[1];        // 1600000

    float* out = (float*)d_out;
    float* ws = (float*)d_ws;
    float* dinv = ws;                            // N
    float* H    = dinv + n_nodes;                // N x 64
    float* F1   = H  + (size_t)n_nodes * 64;     // N x 64
    float* F2   = F1 + (size_t)n_nodes * 64;     // N x 64
    float* AGG  = F2 + (size_t)n_nodes * 64;     // N x 64

    const int TB = 256;
    const int nf = n_nodes * 64;
    const int n_tiles = (n_nodes + 15) / 16;
    const int gemm_blocks = (n_tiles + 7) / 8;  // 8 waves per block

    // degrees -> dinv
    zero_f32<<<(n_nodes + TB - 1) / TB, TB, 0, stream>>>(dinv, n_nodes);
    deg_kernel<<<(n_edges + TB - 1) / TB, TB, 0, stream>>>(dst, dinv, n_edges);
    dinv_kernel<<<(n_nodes + TB - 1) / TB, TB, 0, stream>>>(dinv, n_nodes);

    // H = relu(X @ W1 + b1); H = relu(H @ W2 + b2) in place (row-local dependency)
    gemm_relu_wmma<128, false><<<gemm_blocks, TB, 0, stream>>>(in_feat, nullptr, W1, b1, H, n_nodes);
    gemm_relu_wmma<64, false><<<gemm_blocks, TB, 0, stream>>>(H, nullptr, W2, b2, H, n_nodes);

    // Propagation 1: F1 = H - dinv * A(H * dinv)
    zero_f32<<<(nf + TB - 1) / TB, TB, 0, stream>>>(AGG, nf);
    scatter_kernel<<<(n_edges * 16 + TB - 1) / TB, TB, 0, stream>>>(src, dst, H, dinv, AGG, n_edges);
    update_kernel<<<(n_nodes * 16 + TB - 1) / TB, TB, 0, stream>>>(H, AGG, dinv, F1, n_nodes);

    // Propagation 2: F2 = F1 - dinv * A(F1 * dinv)
    zero_f32<<<(nf + TB - 1) / TB, TB, 0, stream>>>(AGG, nf);
    scatter_kernel<<<(n_edges * 16 + TB - 1) / TB, TB, 0, stream>>>(src, dst, F1, dinv, AGG, n_edges);
    update_kernel<<<(n_nodes * 16 + TB - 1) / TB, TB, 0, stream>>>(F1, AGG, dinv, F2, n_nodes);

    // theta combine (in place): H->out0, F1->out1, F2->out2
    combine_kernel<<<(nf + TB - 1) / TB, TB, 0, stream>>>(H, F1, F2, nf);

    // h_l = relu(out0 @ W3 + b3); h_h = relu([out1 | out2] @ W4 + b4)
    gemm_relu_wmma<64, false><<<gemm_blocks, TB, 0, stream>>>(H, nullptr, W3, b3, out, n_nodes);
    gemm_relu_wmma<128, true><<<gemm_blocks, TB, 0, stream>>>(F1, F2, W4, b4,
                                                              out + (size_t)n_nodes * 64, n_nodes);
}